// AdaptiveSpectrumLayer_83820581748938
// MI455X (gfx1250) — compile-verified
//
#include <hip/hip_runtime.h>
#include <math.h>

// MI455X (gfx1250) implementation of AdaptiveSpectrumLayer.
// B=128, H=T=512, F=16, NF=257, HID=64, K_gate=16448.
// fp32 matrix work on V_WMMA_F32_16X16X4_F32; operand streaming via
// GLOBAL_LOAD_ASYNC_TO_LDS_B128 double-buffering (ASYNCcnt pipeline).

typedef __attribute__((ext_vector_type(2))) float v2f;
typedef __attribute__((ext_vector_type(8))) float v8f;

#define NB   128
#define TT   512
#define FF   16
#define NF   257
#define HID  64

#define C2PI512 0.01227184630308513f   /* 2*pi/512 */
#define S0      0.04419417382415922f   /* 1/sqrt(512) */

__device__ __forceinline__ v8f wmma_f32(v2f a, v2f b, v8f c) {
    return __builtin_amdgcn_wmma_f32_16x16x4_f32(false, a, false, b, (short)0, c, false, false);
}

// low 32 bits of a flat pointer to __shared__ == LDS byte offset
__device__ __forceinline__ unsigned lds_off(const void* p) {
    return (unsigned)(uintptr_t)p;
}
// async 16B global->LDS copy (per active lane), tracked on ASYNCcnt
__device__ __forceinline__ void async_b128(unsigned lds, const void* g) {
    asm volatile("global_load_async_to_lds_b128 %0, %1, off"
                 :: "v"(lds), "v"((unsigned long long)(uintptr_t)g)
                 : "memory");
}
__device__ __forceinline__ void async_wait0()  { asm volatile("s_wait_asynccnt 0"  ::: "memory"); }
__device__ __forceinline__ void async_wait17() { asm volatile("s_wait_asynccnt 17" ::: "memory"); }

// ---------------------------------------------------------------------------
// K0: pad W_gate rows 257 -> 260 floats (16B-aligned rows, zero tail) so the
// gate GEMM can stream it with aligned async b128 copies.
// ---------------------------------------------------------------------------
__global__ __launch_bounds__(256) void k_pack(const float* __restrict__ Wg,
                                              float* __restrict__ Wgp) {
    const int i = blockIdx.x * 256 + threadIdx.x;
    if (i >= 16448 * 260) return;
    const int k = i / 260, c = i % 260;
    Wgp[i] = (c < NF) ? Wg[(size_t)k * NF + c] : 0.f;
}

// ---------------------------------------------------------------------------
// K1: CSD. mean over F^2 cross-corr pairs == (1/F^2) * autocorr of channel sum.
// Then 257-bin ortho DFT of the lag series (lags -255..256 at t=tau+255).
// ---------------------------------------------------------------------------
__global__ __launch_bounds__(256) void k_csd(const float* __restrict__ x,
                                             float* __restrict__ csd_re,
                                             float* __restrict__ csd_im) {
    __shared__ float s[TT];
    __shared__ float c[TT];
    __shared__ float tab[TT];
    const int b = blockIdx.x, tid = threadIdx.x;
    for (int t = tid; t < TT; t += 256) {
        const float* xr = x + ((size_t)b * TT + t) * FF;
        float acc = 0.f;
#pragma unroll
        for (int f = 0; f < FF; ++f) acc += xr[f];
        s[t] = acc;
        tab[t] = cosf(t * C2PI512);
    }
    __syncthreads();
    for (int t = tid; t < TT; t += 256) {
        const int tau = t - 255;
        const int klo = tau < 0 ? -tau : 0;
        const int khi = tau > 0 ? TT - tau : TT;
        float acc = 0.f;
        for (int k = klo; k < khi; ++k) acc += s[k] * s[k + tau];
        c[t] = acc;
    }
    __syncthreads();
    const float s1 = S0 / 256.0f;  // ortho scale * 1/F^2
    for (int n = tid; n < NF; n += 256) {
        float ar = 0.f, ai = 0.f;
        for (int t = 0; t < TT; ++t) {
            const int idx = (n * t) & 511;
            ar += c[t] * tab[idx];
            ai += c[t] * tab[(idx + 128) & 511];   // -sin
        }
        csd_re[b * NF + n] = s1 * ar;
        csd_im[b * NF + n] = s1 * ai;
    }
}

// ---------------------------------------------------------------------------
// K2: forward rfft (ortho) as WMMA GEMM: (272 x 512 DFT) x (512 x 16) per batch,
// + broadcast CSD add. A from LDS cosine table; B async-staged in LDS.
// ---------------------------------------------------------------------------
__global__ __launch_bounds__(256) void k_dft(const float* __restrict__ x,
                                             const float* __restrict__ csd_re,
                                             const float* __restrict__ csd_im,
                                             float* __restrict__ fre,
                                             float* __restrict__ fim) {
    __shared__ float xs[TT * FF];
    __shared__ float tab[TT];
    const int b = blockIdx.x, tid = threadIdx.x;
    {   // async stage x[b] : 8192 floats = 2048 aligned 16B chunks
        const float* src = x + (size_t)b * TT * FF;
        const unsigned base = lds_off(xs);
        for (int c = tid; c < 2048; c += 256)
            async_b128(base + (unsigned)c * 16u, src + (size_t)c * 4);
    }
    for (int i = tid; i < TT; i += 256) tab[i] = cosf(i * C2PI512);
    async_wait0();
    __syncthreads();
    const int lane = tid & 31, wid = tid >> 5;
    const int l15 = lane & 15;
    const int khalf = (lane >> 4) << 1;   // K offset 0 (lanes 0-15) / 2 (16-31)
    const int radd = (lane >> 4) << 3;
    for (int task = wid; task < 34; task += 8) {      // 17 m-tiles x {re,im}
        const int mtile = task >> 1;
        const int off = (task & 1) ? 128 : 0;          // im uses -sin = cos(+pi/2)
        const int nrow = mtile * 16 + l15;
        v8f acc = {};
        for (int ks = 0; ks < 128; ++ks) {
            const int t0 = ks * 4 + khalf;
            v2f a, bb;
            a.x = tab[(nrow * t0 + off) & 511];
            a.y = tab[(nrow * (t0 + 1) + off) & 511];
            bb.x = xs[t0 * FF + l15];
            bb.y = xs[(t0 + 1) * FF + l15];
            acc = wmma_f32(a, bb, acc);
        }
        const float* csd = (task & 1) ? csd_im : csd_re;
        float* out = (task & 1) ? fim : fre;
#pragma unroll
        for (int r = 0; r < 8; ++r) {
            const int n = mtile * 16 + r + radd;
            if (n < NF)
                out[((size_t)b * NF + n) * FF + l15] = S0 * acc[r] + csd[b * NF + n];
        }
    }
}

// ---------------------------------------------------------------------------
// K3: features mag / sin(angle) / cos(angle). angle(0)=0 -> (0,1).
// ---------------------------------------------------------------------------
__global__ __launch_bounds__(256) void k_feat(const float* __restrict__ fre,
                                              const float* __restrict__ fim,
                                              float* __restrict__ mg,
                                              float* __restrict__ si,
                                              float* __restrict__ co, int total) {
    const int i = blockIdx.x * 256 + threadIdx.x;
    if (i >= total) return;
    const float re = fre[i], im = fim[i];
    const float m = sqrtf(re * re + im * im);
    float s, c;
    if (m > 0.f) { s = im / m; c = re / m; } else { s = 0.f; c = 1.f; }
    mg[i] = m; si[i] = s; co[i] = c;
}

// ---------------------------------------------------------------------------
// K4: gate GEMM (M=16/batch, K=16448, N=257) on f32 WMMA.
//  - A (=proj) generated on the fly in LDS from rank-4 features (never in HBM)
//  - B (=W_gate 64x260 slice per freq) streamed global->LDS via async b128,
//    double-buffered: copy of slice n+1 overlaps all compute on slice n.
//  - complexifier heads fused while proj tile resident; SiLU gate in epilogue.
// ---------------------------------------------------------------------------
__global__ __launch_bounds__(256) void k_gate(const float* __restrict__ mg,
                                              const float* __restrict__ si,
                                              const float* __restrict__ co,
                                              const float* __restrict__ Wp,
                                              const float* __restrict__ bp,
                                              const float* __restrict__ Wgp,
                                              const float* __restrict__ bg,
                                              const float* __restrict__ mw,
                                              const float* __restrict__ mb,
                                              const float* __restrict__ pw,
                                              const float* __restrict__ pb,
                                              float* __restrict__ wv,
                                              float* __restrict__ nfr,
                                              float* __restrict__ nfi) {
    __shared__ float wg_s[2][64][264];   // stride 264: conflict-free halves
    __shared__ float proj[2][16 * 68];   // padded stride 68
    __shared__ float wp_s[4 * HID];
    __shared__ float bp_s[HID];
    const int b = blockIdx.x, tid = threadIdx.x;
    const int lane = tid & 31, wid = tid >> 5;
    const int l15 = lane & 15;
    const int khalf = (lane >> 4) << 1;
    if (tid < 256) wp_s[tid] = Wp[tid];
    if (tid < HID) bp_s[tid] = bp[tid];

    // async stream one 64x260 W_gate slice into wg_s[buf]; exactly 17 issues
    // per thread (tail duplicates item 0) so per-wave ASYNCcnt is uniform.
    auto prefetch = [&](int n, int buf) {
        if (n > 256) return;
        const float* src = Wgp + (size_t)n * (64 * 260);
        const unsigned base = lds_off(&wg_s[buf][0][0]);
#pragma unroll 1
        for (int j = 0; j < 17; ++j) {
            int item = tid + j * 256;
            if (item >= 64 * 65) item = 0;          // benign duplicate
            const int r = item / 65, c = item % 65; // c = 16B chunk
            async_b128(base + (unsigned)(r * 264 + c * 4) * 4u,
                       src + (size_t)r * 260 + (size_t)c * 4);
        }
    };
    // 16(f) x 64(h) proj tile for frequency n -> proj[buf]
    auto computeA = [&](int n, int buf) {
        if (n > 256) return;
        const size_t base = ((size_t)b * NF + n) * FF;
        const float freq = n * 0.1953125f;            // n * FS/H
        const int h = tid & 63, fq = tid >> 6;
        const float w0 = wp_s[h], w1 = wp_s[64 + h], w2 = wp_s[128 + h],
                    w3 = wp_s[192 + h], bb = bp_s[h];
#pragma unroll
        for (int j = 0; j < 4; ++j) {
            const int f = fq * 4 + j;
            const float v = mg[base + f] * w0 + si[base + f] * w1 +
                            co[base + f] * w2 + freq * w3 + bb;
            proj[buf][f * 68 + h] = fmaxf(v, 0.f);
        }
    };
    // mag/phase heads: per-f dot over 64 h, reduced across 16-lane groups
    auto complexify = [&](int n, int buf) {
        const int f = tid >> 4, cc = tid & 15;
        float pm = 0.f, pp = 0.f;
#pragma unroll
        for (int j = 0; j < 4; ++j) {
            const float pv = proj[buf][f * 68 + cc + 16 * j];
            pm += pv * mw[n * HID + cc + 16 * j];
            pp += pv * pw[n * HID + cc + 16 * j];
        }
#pragma unroll
        for (int m = 1; m < 16; m <<= 1) {
            pm += __shfl_xor(pm, m, 16);
            pp += __shfl_xor(pp, m, 16);
        }
        if (cc == 0) {
            const float mmag = fmaxf(pm + mb[n], 0.f);
            const float ph = 6.283185307179586f * (1.f / (1.f + expf(-(pp + pb[n]))));
            const size_t o = ((size_t)b * NF + n) * FF + f;
            nfr[o] = mmag * cosf(ph);
            nfi[o] = mmag * sinf(ph);
        }
    };

    v8f acc0 = {}, acc1 = {}, acc2 = {};
    const int ct0 = wid, ct1 = wid + 8;  // ct2=16 handled by wave 0 only
    prefetch(0, 0);
    prefetch(1, 1);
    __syncthreads();          // wp_s/bp_s visible
    computeA(0, 0);
    async_wait17();           // slice 0 landed (only slice 1 in flight)
    __syncthreads();
    for (int n = 0; n < NF; ++n) {
        const int buf = n & 1;
        for (int kk = 0; kk < 16; ++kk) {
            v2f a;
            a.x = proj[buf][l15 * 68 + kk * 4 + khalf];
            a.y = proj[buf][l15 * 68 + kk * 4 + khalf + 1];
            const int row = kk * 4 + khalf;
            {   // ct0: cols 0..127
                v2f bf;
                bf.x = wg_s[buf][row][ct0 * 16 + l15];
                bf.y = wg_s[buf][row + 1][ct0 * 16 + l15];
                acc0 = wmma_f32(a, bf, acc0);
            }
            {   // ct1: cols 128..255
                v2f bf;
                bf.x = wg_s[buf][row][ct1 * 16 + l15];
                bf.y = wg_s[buf][row + 1][ct1 * 16 + l15];
                acc1 = wmma_f32(a, bf, acc1);
            }
            if (wid == 0) {   // ct2=16: only col 256 valid (wave-uniform)
                const int col = 256 + l15;
                v2f bf;
                bf.x = (col < NF) ? wg_s[buf][row][col] : 0.f;
                bf.y = (col < NF) ? wg_s[buf][row + 1][col] : 0.f;
                acc2 = wmma_f32(a, bf, acc2);
            }
        }
        complexify(n, buf);
        computeA(n + 1, buf ^ 1);
        __syncthreads();            // all reads of wg_s[buf]/proj[buf] done
        prefetch(n + 2, buf);       // overwrite consumed buffer
        if (n + 2 <= 256) async_wait17();   // slice n+1 landed
        else              async_wait0();
        __syncthreads();            // wg_s[buf^1] + proj[buf^1] visible
    }

    const int radd = (lane >> 4) << 3;
    auto epi = [&](v8f acc, int ct) {
        const int col = ct * 16 + l15;
#pragma unroll
        for (int r = 0; r < 8; ++r) {
            if (col < NF) {
                const int f = r + radd;
                const float g = acc[r] + bg[col];
                const float sg = g * (1.f / (1.f + expf(-g)));   // SiLU
                const float w = 1.f / (1.f + expf(-sg));
                wv[((size_t)b * NF + col) * FF + f] = w;
            }
        }
    };
    epi(acc0, ct0);
    epi(acc1, ct1);
    if (wid == 0) epi(acc2, 16);
}

// ---------------------------------------------------------------------------
// K5: spectral mix + irfft prescale (ortho scale and the 2x factor for
// bins 1..255 folded into wf so the inverse GEMM uses plain cos/-sin).
// ---------------------------------------------------------------------------
__global__ __launch_bounds__(256) void k_mix(const float* __restrict__ wv,
                                             const float* __restrict__ nfr,
                                             const float* __restrict__ nfi,
                                             const float* __restrict__ fre,
                                             const float* __restrict__ fim,
                                             float* __restrict__ wfr,
                                             float* __restrict__ wfi, int total) {
    const int i = blockIdx.x * 256 + threadIdx.x;
    if (i >= total) return;
    const int n = (i >> 4) % NF;
    const float w = wv[i];
    const float re = w * nfr[i] + (1.f - w) * fre[i];
    const float im = w * nfi[i] + (1.f - w) * fim[i];
    const float pre = ((n == 0 || n == 256) ? 1.f : 2.f) * S0;
    wfr[i] = pre * re;
    wfi[i] = pre * im;
}

// ---------------------------------------------------------------------------
// K6: irfft (ortho) as WMMA GEMM (512 x 260pad) x (260 x 16) per batch,
// async-staged wf, fused residual-add + LayerNorm over F=16 via shuffles.
// ---------------------------------------------------------------------------
__global__ __launch_bounds__(256) void k_irdft(const float* __restrict__ wfr,
                                               const float* __restrict__ wfi,
                                               const float* __restrict__ x,
                                               const float* __restrict__ gamma,
                                               const float* __restrict__ beta,
                                               float* __restrict__ out) {
    __shared__ float wr[260 * FF];
    __shared__ float wi[260 * FF];
    __shared__ float tab[TT];
    const int b = blockIdx.x, tid = threadIdx.x;
    {   // async stage: 4112 floats each = 1028 aligned 16B chunks
        const float* sr = wfr + (size_t)b * NF * FF;
        const float* sj = wfi + (size_t)b * NF * FF;
        const unsigned br_ = lds_off(wr), bi_ = lds_off(wi);
        for (int c = tid; c < 1028; c += 256) {
            async_b128(br_ + (unsigned)c * 16u, sr + (size_t)c * 4);
            async_b128(bi_ + (unsigned)c * 16u, sj + (size_t)c * 4);
        }
    }
    for (int i = NF * FF + tid; i < 260 * FF; i += 256) { wr[i] = 0.f; wi[i] = 0.f; }
    for (int i = tid; i < TT; i += 256) tab[i] = cosf(i * C2PI512);
    async_wait0();
    __syncthreads();
    const int lane = tid & 31, wid = tid >> 5;
    const int l15 = lane & 15;
    const int khalf = (lane >> 4) << 1;
    const int radd = (lane >> 4) << 3;
    const float gam = gamma[l15], bet = beta[l15];
    for (int mt = wid; mt < 32; mt += 8) {
        const int trow = mt * 16 + l15;
        v8f acc = {};
        for (int ks = 0; ks < 65; ++ks) {     // K = 260 (pad rows are zero)
            const int k0 = ks * 4 + khalf;
            v2f ar, ai, br, bi;
            ar.x = tab[(trow * k0) & 511];
            ar.y = tab[(trow * (k0 + 1)) & 511];
            ai.x = tab[(trow * k0 + 128) & 511];        // -sin
            ai.y = tab[(trow * (k0 + 1) + 128) & 511];
            br.x = wr[k0 * FF + l15];
            br.y = wr[(k0 + 1) * FF + l15];
            bi.x = wi[k0 * FF + l15];
            bi.y = wi[(k0 + 1) * FF + l15];
            acc = wmma_f32(ar, br, acc);
            acc = wmma_f32(ai, bi, acc);
        }
#pragma unroll
        for (int r = 0; r < 8; ++r) {
            const int t = mt * 16 + r + radd;
            const size_t idx = ((size_t)b * TT + t) * FF + l15;
            const float y = acc[r] + x[idx];
            float s = y;
            s += __shfl_xor(s, 1, 16); s += __shfl_xor(s, 2, 16);
            s += __shfl_xor(s, 4, 16); s += __shfl_xor(s, 8, 16);
            const float mean = s * 0.0625f;
            const float d = y - mean;
            float q = d * d;
            q += __shfl_xor(q, 1, 16); q += __shfl_xor(q, 2, 16);
            q += __shfl_xor(q, 4, 16); q += __shfl_xor(q, 8, 16);
            const float var = q * 0.0625f;
            out[idx] = d * rsqrtf(var + 1e-5f) * gam + bet;
        }
    }
}

// ---------------------------------------------------------------------------
extern "C" void kernel_launch(void* const* d_in, const int* in_sizes, int n_in,
                              void* d_out, int out_size, void* d_ws, size_t ws_size,
                              hipStream_t stream) {
    (void)in_sizes; (void)n_in; (void)out_size; (void)ws_size;
    const float* x   = (const float*)d_in[0];
    const float* Wp  = (const float*)d_in[1];
    const float* bp  = (const float*)d_in[2];
    const float* Wg  = (const float*)d_in[3];
    const float* bg  = (const float*)d_in[4];
    const float* mw  = (const float*)d_in[5];
    const float* mb  = (const float*)d_in[6];
    const float* pw  = (const float*)d_in[7];
    const float* pb  = (const float*)d_in[8];
    const float* gam = (const float*)d_in[9];
    const float* bet = (const float*)d_in[10];

    float* p = (float*)d_ws;
    const int BNF = NB * NF * FF;            // 526336
    float* csd_re = p; p += NB * NF;
    float* csd_im = p; p += NB * NF;
    float* fre = p; p += BNF;
    float* fim = p; p += BNF;
    float* mg  = p; p += BNF;
    float* si  = p; p += BNF;
    float* co  = p; p += BNF;
    float* wv  = p; p += BNF;
    float* nfr = p; p += BNF;
    float* nfi = p; p += BNF;
    float* wfr = p; p += BNF;
    float* wfi = p; p += BNF;
    float* Wgp = p; p += 16448 * 260;        // padded W_gate
    float* out = (float*)d_out;

    k_pack<<<(16448 * 260 + 255) / 256, 256, 0, stream>>>(Wg, Wgp);
    k_csd<<<NB, 256, 0, stream>>>(x, csd_re, csd_im);
    k_dft<<<NB, 256, 0, stream>>>(x, csd_re, csd_im, fre, fim);
    k_feat<<<(BNF + 255) / 256, 256, 0, stream>>>(fre, fim, mg, si, co, BNF);
    k_gate<<<NB, 256, 0, stream>>>(mg, si, co, Wp, bp, Wgp, bg, mw, mb, pw, pb,
                                   wv, nfr, nfi);
    k_mix<<<(BNF + 255) / 256, 256, 0, stream>>>(wv, nfr, nfi, fre, fim, wfr, wfi, BNF);
    k_irdft<<<NB, 256, 0, stream>>>(wfr, wfi, x, gam, bet, out);
}